// KeyformerPreNormAttn_9878424781193
// MI455X (gfx1250) — compile-verified
//
#include <hip/hip_runtime.h>
#include <hip/hip_bf16.h>

typedef float v2f __attribute__((ext_vector_type(2)));
typedef float v8f __attribute__((ext_vector_type(8)));

#define WMMA_F32_16x16x4(acc, a, b) \
  (acc) = __builtin_amdgcn_wmma_f32_16x16x4_f32(false, (a), false, (b), (short)0, (acc), false, false)

// Problem constants
#define MM 16
#define HH 32
#define DD 128
#define PP 8192
#define NN 4096
#define PT (PP + MM)      // 8208
#define PTILES (PT / 16)  // 513

// ---------------------------------------------------------------------------
// K1: RMS norm; store x_norm in WMMA A-operand permuted order:
//   aperm[g*64 + (m + 16*(j>>1))*2 + (j&1)] = xn[m][4g + j]
// so that lane l's A fragment for k0=4g is a contiguous b64 at aperm+g*64+l*2.
// ---------------------------------------------------------------------------
__global__ void k_rmsnorm(const float* __restrict__ X, float* __restrict__ aperm) {
  int m = blockIdx.x;      // 0..15
  int t = threadIdx.x;     // 0..255
  __shared__ float red[256];
  const float* row = X + (size_t)m * NN;
  float s = 0.f;
  for (int k = t; k < NN; k += 256) { float x = row[k]; s += x * x; }
  red[t] = s; __syncthreads();
  for (int w = 128; w > 0; w >>= 1) { if (t < w) red[t] += red[t + w]; __syncthreads(); }
  float rinv = rsqrtf(red[0] * (1.0f / (float)NN));
  for (int k = t; k < NN; k += 256) {
    int g = k >> 2, j = k & 3;
    aperm[g * 64 + (m + 16 * (j >> 1)) * 2 + (j & 1)] = row[k] * rinv;
  }
}

// ---------------------------------------------------------------------------
// K2: QKV GEMM, split-K=4.  C[m][j] = sum_k xn[m][k] * W[j][k]  (y = x @ W^T)
// One wave per (matrix, jtile, kchunk).  B fragment: lane l reads
// W[jt*16 + l%16][k0 + (l/16)*2 + {0,1}]  -> one b64 per lane per step.
// ---------------------------------------------------------------------------
__global__ __launch_bounds__(32) void k_qkv(const float* __restrict__ aperm,
                                            const float* __restrict__ Wq,
                                            const float* __restrict__ Wk,
                                            const float* __restrict__ Wv,
                                            float* __restrict__ part) {
  int bid = blockIdx.x;          // mat*1024 + jt*4 + ch
  int ch  = bid & 3;
  int jt  = (bid >> 2) & 255;
  int mat = bid >> 10;           // 0..2
  const float* W = (mat == 0) ? Wq : (mat == 1 ? Wk : Wv);
  int l = threadIdx.x;
  const float* arow = aperm + (size_t)(ch * 256) * 64 + l * 2;
  const float* brow = W + (size_t)(jt * 16 + (l & 15)) * NN + ch * 1024 + ((l >> 4) << 1);
  v8f acc = {0.f,0.f,0.f,0.f,0.f,0.f,0.f,0.f};
  for (int o = 0; o < 32; ++o) {
    __builtin_prefetch(brow + o * 32 + 64, 0, 1);   // stream weight rows ahead
#pragma unroll
    for (int u = 0; u < 8; ++u) {
      int it = o * 8 + u;
      v2f a = *(const v2f*)(arow + it * 64);
      v2f b = *(const v2f*)(brow + it * 4);
      WMMA_F32_16x16x4(acc, a, b);
    }
  }
  float* dst = part + (size_t)((mat * 256 + jt) * 4 + ch) * 256 + l;
#pragma unroll
  for (int r = 0; r < 8; ++r) dst[r * 32] = acc[r];
}

// ---------------------------------------------------------------------------
// K2b: reduce 4 K-chunks. q stored in WMMA-perm order (A of score GEMM),
// k/v stored plain [m][4096] (appended B rows of score / output GEMMs).
// ---------------------------------------------------------------------------
__global__ void k_qkv_reduce(const float* __restrict__ part,
                             float* __restrict__ qperm,
                             float* __restrict__ kws,
                             float* __restrict__ vws) {
  int tile = blockIdx.x;        // mat*256 + jt
  int e = threadIdx.x;          // 0..255
  const float* p = part + (size_t)tile * 1024 + e;
  float v = p[0] + p[256] + p[512] + p[768];
  int l = e & 31, r = e >> 5;
  int Mr = r + 8 * (l >> 4), Nc = l & 15;
  int mat = tile >> 8, jt = tile & 255;
  int j = jt * 16 + Nc;
  if (mat == 0) {
    int g = j >> 2, jj = j & 3;
    qperm[g * 64 + (Mr + 16 * (jj >> 1)) * 2 + (jj & 1)] = v;
  } else {
    float* o = (mat == 1) ? kws : vws;
    o[(size_t)Mr * NN + j] = v;
  }
}

// ---------------------------------------------------------------------------
// K3: scores c[h][m][p] = sum_d q[h][m][d]*k_cache[h][p][d].
// One wave per (head, ptile).  ptile 512 is exactly the 16 appended k rows.
// ---------------------------------------------------------------------------
__global__ __launch_bounds__(32) void k_scores(const float* __restrict__ qperm,
                                               const float* __restrict__ kws,
                                               const float* __restrict__ cacheK,
                                               float* __restrict__ cws) {
  int bid = blockIdx.x;          // h*513 + pt
  int pt = bid % PTILES;
  int h  = bid / PTILES;
  int l = threadIdx.x;
  const float* arow = qperm + (size_t)(h * (DD >> 2)) * 64 + l * 2;  // k0 = h*128
  const float* brow;
  if (pt < 512)
    brow = cacheK + ((size_t)h * PP + pt * 16 + (l & 15)) * DD + ((l >> 4) << 1);
  else
    brow = kws + (size_t)(l & 15) * NN + h * DD + ((l >> 4) << 1);
  v8f acc = {0.f,0.f,0.f,0.f,0.f,0.f,0.f,0.f};
#pragma unroll 8
  for (int it = 0; it < 32; ++it) {             // K = 128 = 32 * 4
    v2f a = *(const v2f*)(arow + it * 64);
    v2f b = *(const v2f*)(brow + it * 4);
    WMMA_F32_16x16x4(acc, a, b);
  }
  int Nc = l & 15, mbase = 8 * (l >> 4);
#pragma unroll
  for (int r = 0; r < 8; ++r)
    cws[(size_t)(h * MM + mbase + r) * PT + pt * 16 + Nc] = acc[r];
}

// ---------------------------------------------------------------------------
// K4: per-(h,m) sums of exp(c) and exp((c+noise)/tau); store reciprocals.
// ---------------------------------------------------------------------------
__global__ void k_rowsums(const float* __restrict__ cws,
                          const float* __restrict__ noise,
                          const float* __restrict__ tau,
                          float* __restrict__ invs1, float* __restrict__ invs2) {
  int hm = blockIdx.x;          // 0..511
  int t = threadIdx.x;
  float invtau = 1.0f / tau[0];
  const float* c  = cws   + (size_t)hm * PT;
  const float* nz = noise + (size_t)hm * PT;
  float s1 = 0.f, s2 = 0.f;
  for (int p = t; p < PT; p += 256) {
    float cc = c[p];
    s1 += __expf(cc);
    s2 += __expf((cc + nz[p]) * invtau);
  }
  __shared__ float r1[256], r2[256];
  r1[t] = s1; r2[t] = s2; __syncthreads();
  for (int w = 128; w > 0; w >>= 1) {
    if (t < w) { r1[t] += r1[t + w]; r2[t] += r2[t + w]; }
    __syncthreads();
  }
  if (t == 0) { invs1[hm] = 1.0f / r1[0]; invs2[hm] = 1.0f / r2[0]; }
}

// ---------------------------------------------------------------------------
// K5: c_out[h][p] = sum_m exp((c+noise)/tau)/s2  (second output), and
// overwrite c in place with P1 = exp(c)/s1 so K6 is a pure GEMM.
// ---------------------------------------------------------------------------
__global__ void k_cout_p1(float* __restrict__ cws,
                          const float* __restrict__ noise,
                          const float* __restrict__ tau,
                          const float* __restrict__ invs1,
                          const float* __restrict__ invs2,
                          float* __restrict__ outc) {
  int h = blockIdx.y;
  int p = blockIdx.x * 256 + threadIdx.x;
  if (p >= PT) return;
  float invtau = 1.0f / tau[0];
  float acc = 0.f;
#pragma unroll 4
  for (int m = 0; m < MM; ++m) {
    size_t idx = (size_t)(h * MM + m) * PT + p;
    float cc = cws[idx];
    acc += __expf((cc + noise[idx]) * invtau) * invs2[h * MM + m];
    cws[idx] = __expf(cc) * invs1[h * MM + m];
  }
  outc[(size_t)h * PT + p] = acc;
}

// ---------------------------------------------------------------------------
// K6: o[h][m][d] = sum_p P1[h][m][p] * v_cache[h][p][d], split-K=4.
// One wave per (kchunk, head, dtile).  513 WMMA steps per wave.
// ---------------------------------------------------------------------------
__global__ __launch_bounds__(32) void k_out(const float* __restrict__ cws,
                                            const float* __restrict__ cacheV,
                                            const float* __restrict__ vws,
                                            float* __restrict__ opart) {
  int bid = blockIdx.x;          // ch*256 + h*8 + dt
  int dt = bid & 7;
  int h  = (bid >> 3) & 31;
  int ch = bid >> 8;             // 0..3
  int l = threadIdx.x;
  int Nc = l & 15;
  int kphase = (l >> 4) << 1;
  const float* arow   = cws + (size_t)(h * MM + Nc) * PT + kphase;   // P1 row (L2-resident)
  const float* bbaseV = cacheV + (size_t)h * PP * DD + dt * 16 + Nc;
  const float* bbaseW = vws + h * DD + dt * 16 + Nc;
  v8f acc = {0.f,0.f,0.f,0.f,0.f,0.f,0.f,0.f};
  for (int it = 0; it < 513; ++it) {
    int p0 = (ch * 513 + it) * 4;
    v2f a = *(const v2f*)(arow + p0);
    int pr = p0 + kphase;
    v2f b;
    if (p0 < PP) {                                  // wave-uniform branch
      b[0] = bbaseV[(size_t)pr * DD];
      b[1] = bbaseV[(size_t)(pr + 1) * DD];
    } else {
      b[0] = bbaseW[(size_t)(pr - PP) * NN];
      b[1] = bbaseW[(size_t)(pr - PP + 1) * NN];
    }
    WMMA_F32_16x16x4(acc, a, b);
  }
  float* dst = opart + (size_t)bid * 256 + l;
#pragma unroll
  for (int r = 0; r < 8; ++r) dst[r * 32] = acc[r];
}

// ---------------------------------------------------------------------------
// K7: reduce 4 K-chunks of o and scatter to o2[m][h*128+d] (first output).
// ---------------------------------------------------------------------------
__global__ void k_out_reduce(const float* __restrict__ opart, float* __restrict__ out) {
  int tile = blockIdx.x;        // h*8 + dt
  int e = threadIdx.x;
  const float* p = opart + (size_t)tile * 256 + e;
  float v = p[0] + p[256 * 256] + p[2 * 256 * 256] + p[3 * 256 * 256];
  int l = e & 31, r = e >> 5;
  int Mr = r + 8 * (l >> 4), Nc = l & 15;
  int h = tile >> 3, dt = tile & 7;
  out[(size_t)Mr * NN + h * DD + dt * 16 + Nc] = v;
}

// ---------------------------------------------------------------------------
extern "C" void kernel_launch(void* const* d_in, const int* in_sizes, int n_in,
                              void* d_out, int out_size, void* d_ws, size_t ws_size,
                              hipStream_t stream) {
  const float* X      = (const float*)d_in[0];
  const float* Wq     = (const float*)d_in[1];
  const float* Wk     = (const float*)d_in[2];
  const float* Wv     = (const float*)d_in[3];
  const float* cacheK = (const float*)d_in[4];
  const float* cacheV = (const float*)d_in[5];
  const float* tau    = (const float*)d_in[6];
  const float* noise  = (const float*)d_in[7];
  float* out = (float*)d_out;                 // [0,65536): o2 ; [65536,...): c_out

  float* w = (float*)d_ws;
  float* aperm   = w;                 // 65536
  float* qperm   = w + 65536;         // 65536
  float* kws     = w + 131072;        // 65536
  float* vws     = w + 196608;        // 65536
  float* invs1   = w + 262144;        // 512
  float* invs2   = w + 262656;        // 512
  float* qkvpart = w + 263168;        // 786432
  float* opart   = w + 1049600;       // 262144
  float* cws     = w + 1311744;       // 4202496  (~16.8 MB, L2-resident)

  k_rmsnorm<<<16, 256, 0, stream>>>(X, aperm);
  k_qkv<<<3072, 32, 0, stream>>>(aperm, Wq, Wk, Wv, qkvpart);
  k_qkv_reduce<<<768, 256, 0, stream>>>(qkvpart, qperm, kws, vws);
  k_scores<<<HH * PTILES, 32, 0, stream>>>(qperm, kws, cacheK, cws);
  k_rowsums<<<HH * MM, 256, 0, stream>>>(cws, noise, tau, invs1, invs2);
  k_cout_p1<<<dim3((PT + 255) / 256, HH), 256, 0, stream>>>(cws, noise, tau, invs1, invs2,
                                                            out + 65536);
  k_out<<<1024, 32, 0, stream>>>(cws, cacheV, vws, opart);
  k_out_reduce<<<256, 256, 0, stream>>>(opart, out);
}